// Attention_67740224192738
// MI455X (gfx1250) — compile-verified
//
#include <hip/hip_runtime.h>
#include <math.h>

typedef __attribute__((ext_vector_type(2))) float v2f;
typedef __attribute__((ext_vector_type(8))) float v8f;
typedef __attribute__((ext_vector_type(4))) int   i4;
typedef i4 __attribute__((address_space(1)))* gv4p;   // global int4*
typedef i4 __attribute__((address_space(3)))* lv4p;   // LDS int4*

#define Bn 32
#define Sn 8192
#define Dn 128
#define Hn 128
#define CHUNK 128
#define NCHUNK (Sn / CHUNK)   // 64 chunks -> 2048 one-wave blocks
#define NT (CHUNK / 16)       // 8 s-tiles per wave
#define PART_STRIDE 130       // m, l, acc[128]
#define LDSA_PITCH 132        // floats; conflict-free ds_load_b64, 16B-aligned

// CDNA5 async global->LDS path (ASYNCcnt), double-buffered; falls back to
// plain load+ds_store staging if the builtins are not declared.
#if __has_builtin(__builtin_amdgcn_global_load_async_to_lds_b128) && \
    __has_builtin(__builtin_amdgcn_s_wait_asynccnt)
#define USE_ASYNC 1
#else
#define USE_ASYNC 0
#endif

__device__ __forceinline__ float fast_tanh(float x) {
    // tanh(x) = 1 - 2/(exp(2x)+1)
    float e = __expf(2.0f * x);
    return 1.0f - __fdividef(2.0f, e + 1.0f);
}

// Stage one 16x128 f32 context tile into LDS (row pitch LDSA_PITCH).
// One 16-byte transfer per (row, lane): a full 512B row per step, coalesced.
__device__ __forceinline__ void stage_tile(float* dst, const float* src, int lane) {
#if USE_ASYNC
#pragma unroll
    for (int r = 0; r < 16; ++r) {
        __builtin_amdgcn_global_load_async_to_lds_b128(
            (gv4p)(src + (size_t)r * Dn + 4 * lane),
            (lv4p)(dst + r * LDSA_PITCH + 4 * lane),
            0, 0);
    }
#else
#pragma unroll
    for (int r = 0; r < 16; ++r) {
        *(float4*)(dst + r * LDSA_PITCH + 4 * lane) =
            *(const float4*)(src + (size_t)r * Dn + 4 * lane);
    }
#endif
}

__device__ __forceinline__ void wait_stage() {
#if USE_ASYNC
    __builtin_amdgcn_s_wait_asynccnt(0);
#endif
    __syncthreads();
}

// ---------------- Kernel 0: inp = x @ W_in^T + b_in  (tiny GEMV) ----------------
__global__ __launch_bounds__(128) void inp_kernel(const float* __restrict__ x,
                                                  const float* __restrict__ W_in,
                                                  const float* __restrict__ b_in,
                                                  float* __restrict__ inp) {
    const int b = blockIdx.x;
    const int h = threadIdx.x;
    const float4* xr = (const float4*)(x + (size_t)b * Dn);
    const float4* wr = (const float4*)(W_in + (size_t)h * Dn);
    float s = b_in[h];
#pragma unroll
    for (int i = 0; i < Dn / 4; ++i) {
        float4 a = xr[i], w = wr[i];
        s += a.x * w.x + a.y * w.y + a.z * w.z + a.w * w.w;
    }
    inp[(size_t)b * Hn + h] = s;
}

// ---------------- Kernel 1: WMMA ctx-GEMM + att + online softmax partials ----------------
__global__ __launch_bounds__(32) void attn_tile_kernel(
    const float* __restrict__ context,        // [B,S,D]
    const unsigned char* __restrict__ mask,   // [B,S] bool (1 = masked)
    const float* __restrict__ W_ctx,          // [H,D]
    const float* __restrict__ b_ctx,          // [H]
    const float* __restrict__ V,              // [H]
    const float* __restrict__ inp,            // [B,H]
    float* __restrict__ att_out,              // [B,S]
    float* __restrict__ part) {               // [B,NCHUNK,PART_STRIDE]
    const int chunk = blockIdx.x;
    const int b     = blockIdx.y;
    const int lane  = threadIdx.x;   // 0..31 (wave32)
    const int l15   = lane & 15;
    const int hw    = lane >> 4;     // half-wave id

    __shared__ float ldsA[2][16 * LDSA_PITCH];   // double-buffered A tile

    // Per-lane h parameters for the 8 h-tiles (C layout: N = lane&15)
    float inpv[8], bcv[8], vv[8];
#pragma unroll
    for (int ht = 0; ht < 8; ++ht) {
        const int h = ht * 16 + l15;
        inpv[ht] = inp[(size_t)b * Hn + h];
        bcv[ht]  = b_ctx[h];
        vv[ht]   = V[h];
    }

    // Online softmax state (per half-wave: lanes 0-15 own s-rows 0..7 of each tile,
    // lanes 16-31 own rows 8..15)
    float run_m = -__builtin_inff();
    float run_l = 0.0f;
    float accL[8];
#pragma unroll
    for (int ht = 0; ht < 8; ++ht) accL[ht] = 0.0f;

    const int s0 = chunk * CHUNK;
    const float* cbase = context + ((size_t)b * Sn + s0) * Dn;

    // Prologue: stage tile 0
    stage_tile(&ldsA[0][0], cbase, lane);

    for (int t = 0; t < NT; ++t) {
        const int cur = t & 1;
        wait_stage();                              // tile t resident in ldsA[cur]

        // Prefetch tile t+1 into the other buffer; overlaps with WMMA + epilogue
        if (t + 1 < NT)
            stage_tile(&ldsA[cur ^ 1][0], cbase + (size_t)(t + 1) * 16 * Dn, lane);

        // ---- A fragments (16x4 f32 layout: lanes 0-15 K={k0,k0+1}, 16-31 K={k0+2,k0+3}) ----
        const float* abase = &ldsA[cur][0];
        v2f av[32];
#pragma unroll
        for (int ks = 0; ks < 32; ++ks)
            av[ks] = *(const v2f*)(abase + l15 * LDSA_PITCH + 4 * ks + 2 * hw);

        // ---- GEMM: ctx_tile[16s x 128h] = A(16x128) * W_ctx^T, C init = b_ctx ----
        v8f cv[8];
#pragma unroll
        for (int ht = 0; ht < 8; ++ht) {
            v8f c;
#pragma unroll
            for (int j = 0; j < 8; ++j) c[j] = bcv[ht];
            // B fragment (4x16): lane's two values = W_ctx[ht*16 + l15][k0+2*hw .. +1]
            const float* wrow = W_ctx + (size_t)(ht * 16 + l15) * Dn + 2 * hw;
#pragma unroll
            for (int ks = 0; ks < 32; ++ks) {
                v2f bf = *(const v2f*)(wrow + 4 * ks);
                c = __builtin_amdgcn_wmma_f32_16x16x4_f32(
                        false, av[ks], false, bf, (short)0, c, false, false);
            }
            cv[ht] = c;
        }

        // ---- att[s] = sum_h V[h]*tanh(inp[h] + ctx[h,s]); mask; online softmax update ----
        const int sbase = s0 + t * 16;
        float p[8];
        float m_tile = -__builtin_inff();
#pragma unroll
        for (int j = 0; j < 8; ++j) {
            float aj = 0.0f;
#pragma unroll
            for (int ht = 0; ht < 8; ++ht)
                aj += vv[ht] * fast_tanh(inpv[ht] + cv[ht][j]);
            // 16-lane reduction within each half-wave (xor offsets < 16 stay in half)
            aj += __shfl_xor(aj, 1);
            aj += __shfl_xor(aj, 2);
            aj += __shfl_xor(aj, 4);
            aj += __shfl_xor(aj, 8);
            const int s = sbase + hw * 8 + j;
            const bool msk = mask[(size_t)b * Sn + s] != 0;
            if (msk) aj = -__builtin_inff();
            if (l15 == j) att_out[(size_t)b * Sn + s] = aj;   // masked att is an output
            p[j] = aj;
            m_tile = fmaxf(m_tile, aj);
        }

        const float m_new = fmaxf(run_m, m_tile);
        if (m_new != -__builtin_inff()) {
            const float scale = __expf(run_m - m_new);  // exp(-inf)=0 handles first tile
            run_l *= scale;
#pragma unroll
            for (int ht = 0; ht < 8; ++ht) accL[ht] *= scale;
#pragma unroll
            for (int j = 0; j < 8; ++j) {
                const float pj = __expf(p[j] - m_new);  // masked -> exp(-inf)=0
                run_l += pj;
#pragma unroll
                for (int ht = 0; ht < 8; ++ht) accL[ht] += pj * cv[ht][j];
            }
            run_m = m_new;
        }
    }

    // ---- Merge the two half-wave softmax states ----
    const float om = __shfl_xor(run_m, 16);
    const float ol = __shfl_xor(run_l, 16);
    float oacc[8];
#pragma unroll
    for (int ht = 0; ht < 8; ++ht) oacc[ht] = __shfl_xor(accL[ht], 16);

    const float M = fmaxf(run_m, om);
    float Lp = 0.0f;
    if (M != -__builtin_inff()) {
        const float sa = __expf(run_m - M), sb = __expf(om - M);
        Lp = run_l * sa + ol * sb;
#pragma unroll
        for (int ht = 0; ht < 8; ++ht) accL[ht] = accL[ht] * sa + oacc[ht] * sb;
    } else {
#pragma unroll
        for (int ht = 0; ht < 8; ++ht) accL[ht] = 0.0f;
    }

    float* pp = part + (size_t)(b * NCHUNK + chunk) * PART_STRIDE;
    if (lane == 0) { pp[0] = M; pp[1] = Lp; }
    if (hw == 0) {
#pragma unroll
        for (int ht = 0; ht < 8; ++ht) pp[2 + ht * 16 + l15] = accL[ht];
    }
}

// ---------------- Kernel 2: combine chunk partials -> hidden_state ----------------
__global__ __launch_bounds__(128) void combine_kernel(const float* __restrict__ part,
                                                      float* __restrict__ hidden) {
    const int b = blockIdx.x;
    const int h = threadIdx.x;
    __shared__ float sm[NCHUNK], sl[NCHUNK];
    if (h < NCHUNK) {
        const float* pp = part + (size_t)(b * NCHUNK + h) * PART_STRIDE;
        sm[h] = pp[0];
        sl[h] = pp[1];
    }
    __syncthreads();
    float M = -__builtin_inff();
    for (int i = 0; i < NCHUNK; ++i) M = fmaxf(M, sm[i]);
    float L = 0.0f, acc = 0.0f;
    for (int i = 0; i < NCHUNK; ++i) {
        const float w = __expf(sm[i] - M);   // m_i = -inf (all-masked chunk) -> w = 0
        L   += sl[i] * w;
        acc += part[(size_t)(b * NCHUNK + i) * PART_STRIDE + 2 + h] * w;
    }
    hidden[(size_t)b * Hn + h] = acc / L;
}

extern "C" void kernel_launch(void* const* d_in, const int* in_sizes, int n_in,
                              void* d_out, int out_size, void* d_ws, size_t ws_size,
                              hipStream_t stream) {
    (void)in_sizes; (void)n_in; (void)out_size; (void)ws_size;
    const float*         x       = (const float*)d_in[0];
    const float*         context = (const float*)d_in[1];
    const unsigned char* mask    = (const unsigned char*)d_in[2];
    const float*         W_in    = (const float*)d_in[3];
    const float*         b_in    = (const float*)d_in[4];
    const float*         W_ctx   = (const float*)d_in[5];
    const float*         b_ctx   = (const float*)d_in[6];
    const float*         V       = (const float*)d_in[7];

    float* out        = (float*)d_out;
    float* out_hidden = out;             // [B,H] first
    float* out_att    = out + Bn * Hn;   // then [B,S]

    float* ws      = (float*)d_ws;
    float* ws_inp  = ws;                 // B*H floats
    float* ws_part = ws + Bn * Hn;       // B*NCHUNK*PART_STRIDE floats (~1.04 MB)

    inp_kernel<<<Bn, Hn, 0, stream>>>(x, W_in, b_in, ws_inp);
    attn_tile_kernel<<<dim3(NCHUNK, Bn), 32, 0, stream>>>(
        context, mask, W_ctx, b_ctx, V, ws_inp, out_att, ws_part);
    combine_kernel<<<Bn, Hn, 0, stream>>>(ws_part, out_hidden);
}